// VectorQuantizeLayer_L2_90975997264217
// MI455X (gfx1250) — compile-verified
//
#include <hip/hip_runtime.h>
#include <hip/hip_bf16.h>

// Problem constants (from reference): B=8, S=2048, D_IN=768, D_VQ=256, V=8192
#define M_TOT 16384   // B*S rows
#define K1    768     // D_IN
#define DVQ   256     // D_VQ
#define VCODES 8192   // V

typedef __attribute__((ext_vector_type(2))) float v2f;
typedef __attribute__((ext_vector_type(8))) float v8f;
typedef __attribute__((ext_vector_type(4))) int   v4i;

// ---- CDNA5 async global->LDS path (ASYNCcnt), with safe fallback ----------
#ifndef __has_builtin
#define __has_builtin(x) 0
#endif

#if __has_builtin(__builtin_amdgcn_global_load_async_to_lds_b128)
#define USE_ASYNC_LDS 1
#else
#define USE_ASYNC_LDS 0
#endif

#if __has_builtin(__builtin_amdgcn_s_wait_asynccnt)
#define HAVE_WAIT_ASYNC_BUILTIN 1
#else
#define HAVE_WAIT_ASYNC_BUILTIN 0
#endif

#if USE_ASYNC_LDS
#if HAVE_WAIT_ASYNC_BUILTIN
#define WAIT_ASYNC_LE4() __builtin_amdgcn_s_wait_asynccnt(4)
#define WAIT_ASYNC_LE0() __builtin_amdgcn_s_wait_asynccnt(0)
#else
#define WAIT_ASYNC_LE4() asm volatile("s_wait_asynccnt 0x4" ::: "memory")
#define WAIT_ASYNC_LE0() asm volatile("s_wait_asynccnt 0x0" ::: "memory")
#endif
#else
#define WAIT_ASYNC_LE4()
#define WAIT_ASYNC_LE0()
#endif

// Move 16 bytes global -> LDS. Async path: global_load_async_to_lds_b128
// (no VGPR data movement, tracked by ASYNCcnt). Builtin signature (from the
// round-2 diagnostic): (v4i addrspace(1)*, v4i addrspace(3)*, imm, imm).
__device__ __forceinline__ void copy16_g2lds(const float* g, float* l) {
#if USE_ASYNC_LDS
    __builtin_amdgcn_global_load_async_to_lds_b128(
        (__attribute__((address_space(1))) v4i*)(void*)g,
        (__attribute__((address_space(3))) v4i*)(void*)l, 0, 0);
#else
    *(float4*)l = *(const float4*)g;
#endif
}

// ---------------------------------------------------------------------------
// Kernel 0: per-code half squared norms  nrm[v] = 0.5 * |table[v,:]|^2
// ---------------------------------------------------------------------------
__global__ __launch_bounds__(256)
void k_code_norms(const float* __restrict__ table, float* __restrict__ nrm) {
    const int wid  = threadIdx.x >> 5;
    const int lane = threadIdx.x & 31;
    const int v    = blockIdx.x * 8 + wid;           // 1024 blocks * 8 waves = 8192
    const float* row = table + (size_t)v * DVQ;
    float s = 0.0f;
#pragma unroll
    for (int j = 0; j < 8; ++j) {
        float t = row[lane + 32 * j];
        s += t * t;
    }
#pragma unroll
    for (int m = 16; m >= 1; m >>= 1) s += __shfl_xor(s, m, 32);
    if (lane == 0) nrm[v] = 0.5f * s;
}

// ---------------------------------------------------------------------------
// Kernel 1: h = x @ W^T + b   ([16384,768] x [768,256] -> [16384,256], fp32)
// One 16x16 tile per wave; K=768 -> 192 chained v_wmma_f32_16x16x4_f32.
// ---------------------------------------------------------------------------
__global__ __launch_bounds__(256)
void k_gemm1(const float* __restrict__ x, const float* __restrict__ W,
             const float* __restrict__ bias, float* __restrict__ h) {
    const int wid  = threadIdx.x >> 5;
    const int lane = threadIdx.x & 31;
    const int half = lane >> 4;        // 0: K+{0,1}, 1: K+{2,3}
    const int l16  = lane & 15;

    const int tile  = blockIdx.x * 8 + wid;   // 2048 blocks * 8 = 16384 tiles
    const int mtile = tile >> 4;              // 0..1023
    const int ntile = tile & 15;              // 0..15 (waves in block share x rows)

    const int mrow = mtile * 16 + l16;
    const int ncol = ntile * 16 + l16;
    const float* xp = x + (size_t)mrow * K1 + 2 * half;
    const float* wp = W + (size_t)ncol * K1 + 2 * half;

    v8f acc = {};
#pragma unroll 8
    for (int k0 = 0; k0 < K1; k0 += 4) {
        v2f a = *(const v2f*)(xp + k0);
        v2f b = *(const v2f*)(wp + k0);
        acc = __builtin_amdgcn_wmma_f32_16x16x4_f32(false, a, false, b,
                                                    (short)0, acc, false, false);
    }
    // C layout: VGPR j -> row M = j + 8*half, col N = l16 (all 8 share one N)
    const float bi = bias[ncol];
    float* hp = h + (size_t)(mtile * 16 + 8 * half) * DVQ + ncol;
#pragma unroll
    for (int j = 0; j < 8; ++j) hp[(size_t)j * DVQ] = acc[j] + bi;
}

// ---------------------------------------------------------------------------
// Kernel 2: scores S = h @ table^T, fused argmax over V, fused gather.
// score(m,v) = h.c_v - 0.5|c_v|^2 (monotone transform of reference sim).
//
// One 16-row M-tile per wave (A panel: 64 v2f frags resident in VGPRs for the
// whole V loop). 8 waves/block share each 16x256 table tile staged in LDS,
// DOUBLE-BUFFERED via async global->LDS (ASYNCcnt): tile nt+1 streams into
// buf[(nt+1)&1] while the 64-WMMA K-loop runs on buf[nt&1]. Rows padded to
// 260 floats -> conflict-free ds_load_b64 B-fragments (lane l16 -> banks
// 4*l16 .. 4*l16+3).
// ---------------------------------------------------------------------------
#define NT (VCODES / 16)   // 512 N-tiles

__global__ __launch_bounds__(256)
void k_score_argmax(const float* __restrict__ h, const float* __restrict__ table,
                    const float* __restrict__ nrm, float* __restrict__ out) {
    __shared__ float Bt[2][16][260];   // 2 x (16 codes x 256 f32, +4 pad) = 33 KB

    const int tid  = threadIdx.x;
    const int wid  = tid >> 5;
    const int lane = tid & 31;
    const int half = lane >> 4;
    const int l16  = lane & 15;
    const int mtile = blockIdx.x * 8 + wid;   // 128 blocks * 8 waves = 1024 tiles

    // Preload this wave's A panel: h[mtile*16 + l16, :], K split per ISA layout.
    v2f a[64];
    {
        const float* hp = h + (size_t)(mtile * 16 + l16) * DVQ + 2 * half;
#pragma unroll
        for (int s = 0; s < 64; ++s) a[s] = *(const v2f*)(hp + 4 * s);
    }

    float bestV[8];
    int   bestI[8];
#pragma unroll
    for (int j = 0; j < 8; ++j) { bestV[j] = -3.4e38f; bestI[j] = 0; }

    // Stage tile 0 into buffer 0 (4 x 16B per thread = 16 KB per block).
#pragma unroll
    for (int q = 0; q < 4; ++q) {
        int idx4 = tid + 256 * q;          // 0..1023, 64 float4 per row
        int r = idx4 >> 6, c4 = idx4 & 63;
        copy16_g2lds(table + (size_t)r * DVQ + c4 * 4, &Bt[0][r][c4 * 4]);
    }

    for (int nt = 0; nt < NT; ++nt) {
        if (nt + 1 < NT) {
            // Issue next tile into the other buffer, then wait only for the
            // CURRENT tile's 4 loads (async loads complete in order).
#pragma unroll
            for (int q = 0; q < 4; ++q) {
                int idx4 = tid + 256 * q;
                int r = idx4 >> 6, c4 = idx4 & 63;
                copy16_g2lds(table + (size_t)((nt + 1) * 16 + r) * DVQ + c4 * 4,
                             &Bt[(nt + 1) & 1][r][c4 * 4]);
            }
            WAIT_ASYNC_LE4();
        } else {
            WAIT_ASYNC_LE0();
        }
        __syncthreads();   // all waves' LDS writes for tile nt visible

        v8f acc = {};
        const float* bp = &Bt[nt & 1][l16][2 * half];
#pragma unroll
        for (int s = 0; s < 64; ++s) {
            v2f b = *(const v2f*)(bp + 4 * s);
            acc = __builtin_amdgcn_wmma_f32_16x16x4_f32(false, a[s], false, b,
                                                        (short)0, acc, false, false);
        }
        const int   n  = nt * 16 + l16;
        const float nh = nrm[n];
#pragma unroll
        for (int j = 0; j < 8; ++j) {
            float sc = acc[j] - nh;                 // strict '>' keeps earliest n
            if (sc > bestV[j]) { bestV[j] = sc; bestI[j] = n; }
        }
        __syncthreads();   // readers done before buf[(nt+2)&1] is overwritten
    }

    // Reduce over the 16 lanes of each half (rows j / j+8); ties -> lower idx.
#pragma unroll
    for (int m = 1; m <= 8; m <<= 1) {
#pragma unroll
        for (int j = 0; j < 8; ++j) {
            float ov = __shfl_xor(bestV[j], m, 32);
            int   oi = __shfl_xor(bestI[j], m, 32);
            if (ov > bestV[j] || (ov == bestV[j] && oi < bestI[j])) {
                bestV[j] = ov; bestI[j] = oi;
            }
        }
    }

    // Gather: out[row,:] = table[bestI[row],:]  (forward value of vq_code).
    const float4* tab4 = (const float4*)table;
    float4* out4 = (float4*)out;
#pragma unroll
    for (int r = 0; r < 16; ++r) {
        int idx = __shfl(bestI[r & 7], (r < 8) ? 0 : 16, 32);
        size_t orow = (size_t)(mtile * 16 + r) * 64;
        size_t trow = (size_t)idx * 64;
        out4[orow + lane]      = tab4[trow + lane];
        out4[orow + lane + 32] = tab4[trow + lane + 32];
    }
}

// ---------------------------------------------------------------------------
extern "C" void kernel_launch(void* const* d_in, const int* in_sizes, int n_in,
                              void* d_out, int out_size, void* d_ws, size_t ws_size,
                              hipStream_t stream) {
    const float* x     = (const float*)d_in[0];   // [8,2048,768]
    const float* W     = (const float*)d_in[1];   // [256,768]
    const float* bias  = (const float*)d_in[2];   // [256]
    const float* table = (const float*)d_in[3];   // [8192,256]
    float* out = (float*)d_out;                   // [8,2048,256]

    // Workspace: nrm (8192 f32) at 0, h (16384x256 f32 = 16 MB) at +64KB.
    float* nrm = (float*)d_ws;
    float* h   = (float*)((char*)d_ws + (64 << 10));

    k_code_norms  <<<VCODES / 8, 256, 0, stream>>>(table, nrm);
    k_gemm1       <<<(M_TOT / 16) * (DVQ / 16) / 8, 256, 0, stream>>>(x, W, bias, h);
    k_score_argmax<<<(M_TOT / 16) / 8, 256, 0, stream>>>(h, table, nrm, out);
}